// DualAttention_55619826483242
// MI455X (gfx1250) — compile-verified
//
#include <hip/hip_runtime.h>

typedef __attribute__((ext_vector_type(2))) float v2f;
typedef __attribute__((ext_vector_type(8))) float v8f;

static constexpr int Bsz = 4;
static constexpr int C   = 512;
static constexpr int C8  = 64;
static constexpr int N   = 4096;   // 64*64

// D = A(16x4) * B(4x16) + C, fp32 WMMA.
// A frag: lane holds M=lane%16, K = v + 2*(lane/16)
// B frag: lane holds N=lane%16, K = v + 2*(lane/16)
// D frag: vgpr r -> M = r + 8*(lane/16), N = lane%16
__device__ __forceinline__ v8f wmma4(v2f a, v2f b, v8f c) {
  return __builtin_amdgcn_wmma_f32_16x16x4_f32(false, a, false, b, (short)0, c,
                                               false, false);
}

// ---------------------------------------------------------------------------
// Projection GEMM: Out[b,n,d] = sum_c W[d,c] * x[b,c,n] + bias[d]
// x: (B,C,N), W: (D,C), Out: (B,N,D).  grid(N/64, D/64, B), block 128 (4 waves)
// ---------------------------------------------------------------------------
__global__ __launch_bounds__(128)
void proj_kernel(const float* __restrict__ x, const float* __restrict__ W,
                 const float* __restrict__ bias, float* __restrict__ out, int D) {
  const int n0 = blockIdx.x * 64;
  const int d0 = blockIdx.y * 64;
  const int b  = blockIdx.z;
  const int tid = threadIdx.x;
  const int wave = tid >> 5;
  const int lane = tid & 31;
  const int lm = lane & 15, lh = lane >> 4;

  __shared__ float xs[16][65];  // [k(c)][n]
  __shared__ float wsb[64][17]; // [d][k(c)]

  const float* xb = x + (size_t)b * C * N;

  v8f zero = {};
  v8f acc[4];
#pragma unroll
  for (int t = 0; t < 4; ++t) acc[t] = zero;

  for (int k0 = 0; k0 < C; k0 += 16) {
    __syncthreads();
    for (int i = tid; i < 16 * 64; i += 128) {
      int kk = i >> 6, nn = i & 63;
      xs[kk][nn] = xb[(size_t)(k0 + kk) * N + n0 + nn];
    }
    for (int i = tid; i < 64 * 16; i += 128) {
      int dd = i >> 4, kk = i & 15;
      wsb[dd][kk] = W[(size_t)(d0 + dd) * C + k0 + kk];
    }
    __syncthreads();
#pragma unroll
    for (int s = 0; s < 4; ++s) {
      const int kb = 4 * s + 2 * lh;
      v2f a;
      a.x = xs[kb][wave * 16 + lm];
      a.y = xs[kb + 1][wave * 16 + lm];
#pragma unroll
      for (int t = 0; t < 4; ++t) {
        v2f bf;
        bf.x = wsb[t * 16 + lm][kb];
        bf.y = wsb[t * 16 + lm][kb + 1];
        acc[t] = wmma4(a, bf, acc[t]);
      }
    }
  }
#pragma unroll
  for (int t = 0; t < 4; ++t) {
    const int d = d0 + t * 16 + lm;
    const float bv = bias[d];
#pragma unroll
    for (int r = 0; r < 8; ++r) {
      const int n = n0 + wave * 16 + r + 8 * lh;
      out[((size_t)b * N + n) * D + d] = acc[t][r] + bv;
    }
  }
}

// ---------------------------------------------------------------------------
// Flash attention: P[b,n,c] = sum_j softmax_j(Q[n].K[j]) * V[j,c]
// Q,K: (B,N,64), V: (B,N,512), P: (B,N,512)
// grid(N/16, B), block 64 (2 waves; wave w owns V columns [w*256, w*256+256))
// ---------------------------------------------------------------------------
__global__ __launch_bounds__(64, 1)
void flash_attn_kernel(const float* __restrict__ Q, const float* __restrict__ Km,
                       const float* __restrict__ V, float* __restrict__ P) {
  const int q0 = blockIdx.x * 16;
  const int b  = blockIdx.y;
  const int tid = threadIdx.x;
  const int wave = tid >> 5;
  const int lane = tid & 31;
  const int lm = lane & 15, lh = lane >> 4;

  __shared__ float psh[2][16][17];

  const float* Qb = Q + ((size_t)b * N + q0) * C8;
  const float* Kb = Km + (size_t)b * N * C8;
  const float* Vb = V + (size_t)b * N * C;

  // Q fragments (16 k-steps of K=4), resident for the whole sweep
  v2f qf[16];
#pragma unroll
  for (int s = 0; s < 16; ++s) {
    const int k = 4 * s + 2 * lh;
    qf[s].x = Qb[lm * C8 + k];
    qf[s].y = Qb[lm * C8 + k + 1];
  }

  v8f zero = {};
  v8f acc[16];  // 16 tiles of 16x16 -> 256 V columns per wave
#pragma unroll
  for (int t = 0; t < 16; ++t) acc[t] = zero;

  float m8[8], l8[8];
#pragma unroll
  for (int r = 0; r < 8; ++r) { m8[r] = -3.0e38f; l8[r] = 0.0f; }

  for (int jt = 0; jt < N / 16; ++jt) {
    // S = Q(16x64) @ K_tile^T(64x16)
    const float* Kt = Kb + (size_t)jt * 16 * C8;
    v8f s = zero;
#pragma unroll
    for (int ss = 0; ss < 16; ++ss) {
      const int k = 4 * ss + 2 * lh;
      v2f kb;
      kb.x = Kt[lm * C8 + k];
      kb.y = Kt[lm * C8 + k + 1];
      s = wmma4(qf[ss], kb, s);
    }
    // online softmax: row = r + 8*lh, its 16 cols live in one 16-lane half
    float scale[8];
#pragma unroll
    for (int r = 0; r < 8; ++r) {
      const float v = s[r];
      float mx = v;
#pragma unroll
      for (int off = 8; off >= 1; off >>= 1)
        mx = fmaxf(mx, __shfl_xor(mx, off, 32));
      const float mnew = fmaxf(m8[r], mx);
      const float p = __expf(v - mnew);
      float sum = p;
#pragma unroll
      for (int off = 8; off >= 1; off >>= 1)
        sum += __shfl_xor(sum, off, 32);
      const float corr = __expf(m8[r] - mnew);
      l8[r] = l8[r] * corr + sum;
      m8[r] = mnew;
      scale[r] = corr;
      psh[wave][r + 8 * lh][lm] = p;  // D-layout -> LDS
    }
#pragma unroll
    for (int t = 0; t < 16; ++t)
#pragma unroll
      for (int r = 0; r < 8; ++r) acc[t][r] *= scale[r];

    __syncthreads();
    // re-fetch P in A-fragment layout
    v2f pf[4];
#pragma unroll
    for (int ss = 0; ss < 4; ++ss) {
      const int k = 4 * ss + 2 * lh;
      pf[ss].x = psh[wave][lm][k];
      pf[ss].y = psh[wave][lm][k + 1];
    }
    // acc += P(16x16) @ V_tile(16x512-slice)
    const float* Vt = Vb + (size_t)jt * 16 * C;
#pragma unroll
    for (int t = 0; t < 16; ++t) {
      const int cb = (wave * 16 + t) * 16 + lm;
#pragma unroll
      for (int ss = 0; ss < 4; ++ss) {
        const int k = 4 * ss + 2 * lh;
        v2f vb;
        vb.x = Vt[(size_t)k * C + cb];
        vb.y = Vt[(size_t)(k + 1) * C + cb];
        acc[t] = wmma4(pf[ss], vb, acc[t]);
      }
    }
    __syncthreads();
  }

  float inv[8];
#pragma unroll
  for (int r = 0; r < 8; ++r) inv[r] = 1.0f / l8[r];
  float* Pb = P + ((size_t)b * N + q0) * C;
#pragma unroll
  for (int t = 0; t < 16; ++t) {
    const int cb = (wave * 16 + t) * 16 + lm;
#pragma unroll
    for (int r = 0; r < 8; ++r)
      Pb[(size_t)(r + 8 * lh) * C + cb] = acc[t][r] * inv[r];
  }
}

// ---------------------------------------------------------------------------
// Fuse + transpose: P[b,n,c] = gamma*P[b,n,c] + x[b,c,n]  (in place, = pos^T)
//                   pos_cn[b,c,n] = same value (transposed copy)
// grid(N/64, C/64, B), block 256
// ---------------------------------------------------------------------------
__global__ __launch_bounds__(256)
void fuse_kernel(float* __restrict__ P, const float* __restrict__ x,
                 const float* __restrict__ gamma, float* __restrict__ pos_cn) {
  __shared__ float tile[64][65];
  const int n0 = blockIdx.x * 64, c0 = blockIdx.y * 64, b = blockIdx.z;
  const float g = gamma[0];
  const int tid = threadIdx.x;

  for (int i = tid; i < 64 * 64; i += 256) {  // x tile, coalesced in n
    const int cc = i >> 6, nn = i & 63;
    tile[cc][nn] = x[((size_t)(b * C + c0 + cc)) * N + n0 + nn];
  }
  __syncthreads();
  for (int i = tid; i < 64 * 64; i += 256) {  // P coalesced in c; fuse in place
    const int nn = i >> 6, cc = i & 63;
    const size_t pidx = ((size_t)b * N + n0 + nn) * C + c0 + cc;
    const float v = g * P[pidx] + tile[cc][nn];
    P[pidx] = v;
    tile[cc][nn] = v;  // one thread per element: no race
  }
  __syncthreads();
  for (int i = tid; i < 64 * 64; i += 256) {  // transposed copy, coalesced in n
    const int cc = i >> 6, nn = i & 63;
    pos_cn[((size_t)(b * C + c0 + cc)) * N + n0 + nn] = tile[cc][nn];
  }
}

// ---------------------------------------------------------------------------
// ce[b,c,d] = sum_n pos[c,n]*pos[d,n] = (P^T P)[c,d],  P = (N,C)
// grid(C/64, C/64, B), block 128 (4 waves)
// ---------------------------------------------------------------------------
__global__ __launch_bounds__(128)
void ce_kernel(const float* __restrict__ P, float* __restrict__ ce) {
  const int c0 = blockIdx.x * 64, d0 = blockIdx.y * 64, b = blockIdx.z;
  const int tid = threadIdx.x;
  const int wave = tid >> 5, lane = tid & 31;
  const int lm = lane & 15, lh = lane >> 4;

  __shared__ float ta[16][65];  // [k(n)][c]
  __shared__ float tb[16][65];  // [k(n)][d]

  const float* Pb = P + (size_t)b * N * C;
  v8f zero = {};
  v8f acc[4];
#pragma unroll
  for (int t = 0; t < 4; ++t) acc[t] = zero;

  for (int k0 = 0; k0 < N; k0 += 16) {
    __syncthreads();
    for (int i = tid; i < 16 * 64; i += 128) {
      const int kk = i >> 6, cc = i & 63;
      ta[kk][cc] = Pb[(size_t)(k0 + kk) * C + c0 + cc];
      tb[kk][cc] = Pb[(size_t)(k0 + kk) * C + d0 + cc];
    }
    __syncthreads();
#pragma unroll
    for (int s = 0; s < 4; ++s) {
      const int kb = 4 * s + 2 * lh;
      v2f a;
      a.x = ta[kb][wave * 16 + lm];
      a.y = ta[kb + 1][wave * 16 + lm];
#pragma unroll
      for (int t = 0; t < 4; ++t) {
        v2f bf;
        bf.x = tb[kb][t * 16 + lm];
        bf.y = tb[kb + 1][t * 16 + lm];
        acc[t] = wmma4(a, bf, acc[t]);
      }
    }
  }
#pragma unroll
  for (int t = 0; t < 4; ++t) {
    const int d = d0 + t * 16 + lm;
#pragma unroll
    for (int r = 0; r < 8; ++r) {
      const int c = c0 + wave * 16 + r + 8 * lh;
      ce[((size_t)b * C + c) * C + d] = acc[t][r];
    }
  }
}

// ---------------------------------------------------------------------------
// In-place row softmax over 512 columns. grid(B*C), block 32 (one wave/row)
// ---------------------------------------------------------------------------
__global__ __launch_bounds__(32)
void softmax512_kernel(float* __restrict__ ce) {
  float* p = ce + (size_t)blockIdx.x * C;
  const int lane = threadIdx.x;
  float vals[16];
  float mx = -3.0e38f;
#pragma unroll
  for (int i = 0; i < 16; ++i) {
    vals[i] = p[lane + i * 32];
    mx = fmaxf(mx, vals[i]);
  }
#pragma unroll
  for (int off = 16; off >= 1; off >>= 1)
    mx = fmaxf(mx, __shfl_xor(mx, off, 32));
  float sum = 0.0f;
#pragma unroll
  for (int i = 0; i < 16; ++i) {
    vals[i] = __expf(vals[i] - mx);
    sum += vals[i];
  }
#pragma unroll
  for (int off = 16; off >= 1; off >>= 1)
    sum += __shfl_xor(sum, off, 32);
  const float inv = 1.0f / sum;
#pragma unroll
  for (int i = 0; i < 16; ++i) p[lane + i * 32] = vals[i] * inv;
}

// ---------------------------------------------------------------------------
// out[b,c,n] = cg * sum_d cattn[c,d]*pos[d,n] + pos[c,n]
//            = cg * (cattn @ P^T)[c,n] + pos_cn[c,n]
// cattn: (B,C,C), P: (B,N,C), pos_cn: (B,C,N), out: (B,C,N)
// grid(N/64, C/64, B), block 128 (4 waves)
// ---------------------------------------------------------------------------
__global__ __launch_bounds__(128)
void out_kernel(const float* __restrict__ cattn, const float* __restrict__ P,
                const float* __restrict__ pos_cn, const float* __restrict__ cgamma,
                float* __restrict__ out) {
  const int n0 = blockIdx.x * 64, c0 = blockIdx.y * 64, b = blockIdx.z;
  const int tid = threadIdx.x;
  const int wave = tid >> 5, lane = tid & 31;
  const int lm = lane & 15, lh = lane >> 4;

  __shared__ float ta[64][17];  // [c][k(d)]
  __shared__ float tb[64][17];  // [n][k(d)]

  const float* Ab = cattn + (size_t)b * C * C;
  const float* Pb = P + (size_t)b * N * C;
  v8f zero = {};
  v8f acc[4];
#pragma unroll
  for (int t = 0; t < 4; ++t) acc[t] = zero;

  for (int k0 = 0; k0 < C; k0 += 16) {
    __syncthreads();
    for (int i = tid; i < 64 * 16; i += 128) {
      const int rr = i >> 4, kk = i & 15;
      ta[rr][kk] = Ab[(size_t)(c0 + rr) * C + k0 + kk];
      tb[rr][kk] = Pb[(size_t)(n0 + rr) * C + k0 + kk];
    }
    __syncthreads();
#pragma unroll
    for (int s = 0; s < 4; ++s) {
      const int kb = 4 * s + 2 * lh;
      v2f a;
      a.x = ta[wave * 16 + lm][kb];
      a.y = ta[wave * 16 + lm][kb + 1];
#pragma unroll
      for (int t = 0; t < 4; ++t) {
        v2f bf;
        bf.x = tb[t * 16 + lm][kb];
        bf.y = tb[t * 16 + lm][kb + 1];
        acc[t] = wmma4(a, bf, acc[t]);
      }
    }
  }
  const float cg = cgamma[0];
#pragma unroll
  for (int t = 0; t < 4; ++t) {
    const int n = n0 + t * 16 + lm;
#pragma unroll
    for (int r = 0; r < 8; ++r) {
      const int c = c0 + wave * 16 + r + 8 * lh;
      const size_t idx = ((size_t)(b * C + c)) * N + n;
      out[idx] = cg * acc[t][r] + pos_cn[idx];
    }
  }
}

// ---------------------------------------------------------------------------
extern "C" void kernel_launch(void* const* d_in, const int* in_sizes, int n_in,
                              void* d_out, int out_size, void* d_ws, size_t ws_size,
                              hipStream_t stream) {
  (void)in_sizes; (void)n_in; (void)out_size; (void)ws_size;
  const float* x      = (const float*)d_in[0];
  const float* Wq     = (const float*)d_in[1];
  const float* bq     = (const float*)d_in[2];
  const float* Wk     = (const float*)d_in[3];
  const float* bk     = (const float*)d_in[4];
  const float* Wv     = (const float*)d_in[5];
  const float* bv     = (const float*)d_in[6];
  const float* gamma  = (const float*)d_in[7];
  const float* cgamma = (const float*)d_in[8];
  float* out = (float*)d_out;

  float* ws   = (float*)d_ws;
  float* Qb   = ws;                              // B*N*C8
  float* Kb   = Qb + (size_t)Bsz * N * C8;       // B*N*C8
  float* Vb   = Kb + (size_t)Bsz * N * C8;       // B*N*C
  float* Pb   = Vb + (size_t)Bsz * N * C;        // B*N*C  (attn@V, then pos^T)
  float* PosC = Pb + (size_t)Bsz * N * C;        // B*C*N  (pos, channel-major)
  float* CE   = PosC + (size_t)Bsz * C * N;      // B*C*C

  proj_kernel<<<dim3(N / 64, C8 / 64, Bsz), 128, 0, stream>>>(x, Wq, bq, Qb, C8);
  proj_kernel<<<dim3(N / 64, C8 / 64, Bsz), 128, 0, stream>>>(x, Wk, bk, Kb, C8);
  proj_kernel<<<dim3(N / 64, C / 64, Bsz), 128, 0, stream>>>(x, Wv, bv, Vb, C);
  flash_attn_kernel<<<dim3(N / 16, Bsz), 64, 0, stream>>>(Qb, Kb, Vb, Pb);
  fuse_kernel<<<dim3(N / 64, C / 64, Bsz), 256, 0, stream>>>(Pb, x, gamma, PosC);
  ce_kernel<<<dim3(C / 64, C / 64, Bsz), 128, 0, stream>>>(Pb, CE);
  softmax512_kernel<<<dim3(Bsz * C), 32, 0, stream>>>(CE);
  out_kernel<<<dim3(N / 64, C / 64, Bsz), 128, 0, stream>>>(CE, Pb, PosC, cgamma, out);
}